// GCN_6751688589762
// MI455X (gfx1250) — compile-verified
//
#include <hip/hip_runtime.h>
#include <hip/hip_bf16.h>

typedef float v2f __attribute__((ext_vector_type(2)));
typedef float v8f __attribute__((ext_vector_type(8)));

// ---------------------------------------------------------------- degree
__global__ void k_deg_init(float* deg, int n) {
    int i = blockIdx.x * blockDim.x + threadIdx.x;
    if (i < n) deg[i] = 1.0f;   // self-loop
}

__global__ void k_deg_count(const int* __restrict__ dst, float* deg, int e) {
    int i = blockIdx.x * blockDim.x + threadIdx.x;
    if (i < e) atomicAdd(&deg[dst[i]], 1.0f);
}

__global__ void k_rsqrt_inplace(float* deg, int n) {
    int i = blockIdx.x * blockDim.x + threadIdx.x;
    if (i < n) deg[i] = rsqrtf(deg[i]);
}

// ---------------------------------------------------------------- layer 1 (F_in = 1, rank-1 GEMM)
__global__ void k_xw1(const float* __restrict__ x, const float* __restrict__ W1,
                      float* __restrict__ t, int n) {
    int i = blockIdx.x * blockDim.x + threadIdx.x;
    if (i < n * 128) {
        int node = i >> 7;
        int f    = i & 127;
        t[i] = x[node] * W1[f];
    }
}

// ---------------------------------------------------------------- aggregation
// out[d,:] starts with the self-loop contribution dis[d]^2 * t[d,:]
template <int F>
__global__ void k_agg_init(const float* __restrict__ t, const float* __restrict__ dis,
                           float* __restrict__ out, int n) {
    int i = blockIdx.x * blockDim.x + threadIdx.x;
    if (i < n * F) {
        int node = i / F;
        float d = dis[node];
        out[i] = d * d * t[i];
    }
}

// per-edge scatter: out[d,:] += dis[s]*dis[d] * t[s,:]  (4 channels per thread)
template <int F>
__global__ void k_agg_edges(const float* __restrict__ t, const float* __restrict__ dis,
                            const int* __restrict__ src, const int* __restrict__ dst,
                            float* __restrict__ out, int e) {
    constexpr int F4 = F / 4;
    int i = blockIdx.x * blockDim.x + threadIdx.x;
    if (i >= e * F4) return;
    int edge = i / F4;
    int c4   = (i % F4) * 4;
    int s = src[edge];
    int d = dst[edge];
    float nrm = dis[s] * dis[d];
    const float4 v = *(const float4*)(t + (size_t)s * F + c4);
    float* o = out + (size_t)d * F + c4;
    atomicAdd(o + 0, nrm * v.x);
    atomicAdd(o + 1, nrm * v.y);
    atomicAdd(o + 2, nrm * v.z);
    atomicAdd(o + 3, nrm * v.w);
}

template <int F>
__global__ void k_bias_relu(float* __restrict__ out, const float* __restrict__ b, int n) {
    int i = blockIdx.x * blockDim.x + threadIdx.x;
    if (i < n * F) {
        int f = i & (F - 1);
        out[i] = fmaxf(out[i] + b[f], 0.0f);
    }
}

// ---------------------------------------------------------------- f32 WMMA GEMM: C = A[MxK] * B[KxN]
// One wave computes one 16x16 tile. M,N multiples of 16, K multiple of 4.
// Tile id is wave-uniform -> EXEC stays all-ones inside (WMMA requirement).
__global__ void k_gemm_wmma(const float* __restrict__ A, const float* __restrict__ B,
                            float* __restrict__ C, int M, int N, int K) {
    const int lane  = threadIdx.x & 31;
    const int wave  = threadIdx.x >> 5;
    const int nT    = N >> 4;
    const int tile  = blockIdx.x * (blockDim.x >> 5) + wave;
    const int mTile = tile / nT;
    const int nTile = tile % nT;
    if (mTile >= (M >> 4)) return;                 // uniform per wave

    const int half  = lane >> 4;                   // 0: lanes 0-15, 1: lanes 16-31
    const int l     = lane & 15;
    const int mBase = mTile << 4;
    const int nBase = nTile << 4;

    // A (16x4/step): lane half 0 -> K=k0,k0+1 ; half 1 -> K=k0+2,k0+3 ; M = l
    const float* arow = A + (size_t)(mBase + l) * K + 2 * half;
    // B (4x16/step): lane half 0 -> rows k0,k0+1 ; half 1 -> rows k0+2,k0+3 ; N = nBase+l
    const float* bcol = B + (size_t)(2 * half) * N + nBase + l;

    v8f acc = {};
    for (int k0 = 0; k0 < K; k0 += 4) {
        v2f a;
        a.x = arow[k0 + 0];
        a.y = arow[k0 + 1];
        v2f b;
        b.x = bcol[(size_t)k0 * N];
        b.y = bcol[(size_t)(k0 + 1) * N];
        acc = __builtin_amdgcn_wmma_f32_16x16x4_f32(
            /*neg_a=*/false, a, /*neg_b=*/false, b,
            /*c_mod=*/(short)0, acc, /*reuse_a=*/false, /*reuse_b=*/false);
    }

    // D layout: VGPR r -> M = r + 8*half, N = l
#pragma unroll
    for (int r = 0; r < 8; ++r) {
        C[(size_t)(mBase + r + 8 * half) * N + nBase + l] = acc[r];
    }
}

// ---------------------------------------------------------------- global max pool
__global__ void k_pool_init(float* g, int total) {
    int i = blockIdx.x * blockDim.x + threadIdx.x;
    if (i < total) g[i] = 0.0f;    // h3 >= 0 (post-ReLU): bit-max with 0-init == segment_max
}

__global__ void k_pool_max(const float* __restrict__ h, const int* __restrict__ batch,
                           float* __restrict__ g, int n) {
    int i = blockIdx.x * blockDim.x + threadIdx.x;
    if (i < n * 256) {
        int node = i >> 8;
        int f    = i & 255;
        unsigned v = __float_as_uint(h[i]);           // non-negative -> order-preserving bits
        atomicMax((unsigned*)&g[(size_t)batch[node] * 256 + f], v);
    }
}

// ---------------------------------------------------------------- FC head
__global__ void k_fc1(const float* __restrict__ g, const float* __restrict__ W,
                      const float* __restrict__ b, float* __restrict__ o) {
    int i = blockIdx.x * blockDim.x + threadIdx.x;   // 64*128
    if (i >= 64 * 128) return;
    int row = i >> 7, col = i & 127;
    float acc = b[col];
    for (int k = 0; k < 256; ++k) acc += g[row * 256 + k] * W[k * 128 + col];
    o[i] = fmaxf(acc, 0.0f);
}

__global__ void k_fc2(const float* __restrict__ g1, const float* __restrict__ W,
                      const float* __restrict__ b, float* __restrict__ o) {
    int i = blockIdx.x * blockDim.x + threadIdx.x;   // 64*10
    if (i >= 64 * 10) return;
    int row = i / 10, col = i % 10;
    float acc = b[col];
    for (int k = 0; k < 128; ++k) acc += g1[row * 128 + k] * W[k * 10 + col];
    o[i] = fmaxf(acc, 0.0f);
}

// ---------------------------------------------------------------- launch
extern "C" void kernel_launch(void* const* d_in, const int* in_sizes, int n_in,
                              void* d_out, int out_size, void* d_ws, size_t ws_size,
                              hipStream_t stream) {
    const float* x   = (const float*)d_in[0];
    const int*   ei  = (const int*)d_in[1];
    const int*   bat = (const int*)d_in[2];
    const float* W1  = (const float*)d_in[3];
    const float* b1  = (const float*)d_in[4];
    const float* W2  = (const float*)d_in[5];
    const float* b2  = (const float*)d_in[6];
    const float* W3  = (const float*)d_in[7];
    const float* b3  = (const float*)d_in[8];
    const float* Wf1 = (const float*)d_in[9];
    const float* bf1 = (const float*)d_in[10];
    const float* Wf2 = (const float*)d_in[11];
    const float* bf2 = (const float*)d_in[12];

    const int N = in_sizes[0];          // 50000
    const int E = in_sizes[1] / 2;      // 800000
    const int G = 64;
    const int* src = ei;
    const int* dst = ei + E;

    // workspace carving (256B aligned)
    size_t off = 0;
    auto carve = [&](size_t bytes) -> char* {
        char* p = (char*)d_ws + off;
        off += (bytes + 255) & ~(size_t)255;
        return p;
    };
    float* dis  = (float*)carve((size_t)N * 4);          // deg then rsqrt(deg)
    float* bufA = (float*)carve((size_t)N * 128 * 4);    // t (X*W) for layers 1/2
    float* bufB = (float*)carve((size_t)N * 128 * 4);    // h (aggregated) layers 1/2
    float* bufC = (float*)carve((size_t)N * 256 * 4);    // t3
    float* bufD = (float*)carve((size_t)N * 256 * 4);    // h3
    float* gbuf = (float*)carve((size_t)G * 256 * 4);    // pooled
    float* g1   = (float*)carve((size_t)G * 128 * 4);    // fc1 out
    (void)ws_size; (void)n_in; (void)out_size;

    const int T = 256;
    auto blks = [](long total, int t) { return (int)((total + t - 1) / t); };

    // degree + normalization
    k_deg_init<<<blks(N, T), T, 0, stream>>>(dis, N);
    k_deg_count<<<blks(E, T), T, 0, stream>>>(dst, dis, E);
    k_rsqrt_inplace<<<blks(N, T), T, 0, stream>>>(dis, N);

    // ---- layer 1: t1 = x*W1 ; h1 = relu(agg(t1) + b1)
    k_xw1<<<blks((long)N * 128, T), T, 0, stream>>>(x, W1, bufA, N);
    k_agg_init<128><<<blks((long)N * 128, T), T, 0, stream>>>(bufA, dis, bufB, N);
    k_agg_edges<128><<<blks((long)E * 32, T), T, 0, stream>>>(bufA, dis, src, dst, bufB, E);
    k_bias_relu<128><<<blks((long)N * 128, T), T, 0, stream>>>(bufB, b1, N);

    // ---- layer 2: t2 = h1 @ W2 (WMMA) ; h2 = relu(agg(t2) + b2)
    {
        int tiles = (N >> 4) * (128 >> 4);
        k_gemm_wmma<<<blks(tiles, 4), 128, 0, stream>>>(bufB, W2, bufA, N, 128, 128);
    }
    k_agg_init<128><<<blks((long)N * 128, T), T, 0, stream>>>(bufA, dis, bufB, N);
    k_agg_edges<128><<<blks((long)E * 32, T), T, 0, stream>>>(bufA, dis, src, dst, bufB, E);
    k_bias_relu<128><<<blks((long)N * 128, T), T, 0, stream>>>(bufB, b2, N);

    // ---- layer 3: t3 = h2 @ W3 (WMMA) ; h3 = relu(agg(t3) + b3)
    {
        int tiles = (N >> 4) * (256 >> 4);
        k_gemm_wmma<<<blks(tiles, 4), 128, 0, stream>>>(bufB, W3, bufC, N, 256, 128);
    }
    k_agg_init<256><<<blks((long)N * 256, T), T, 0, stream>>>(bufC, dis, bufD, N);
    k_agg_edges<256><<<blks((long)E * 64, T), T, 0, stream>>>(bufC, dis, src, dst, bufD, E);
    k_bias_relu<256><<<blks((long)N * 256, T), T, 0, stream>>>(bufD, b3, N);

    // ---- global max pool over batch
    k_pool_init<<<blks(G * 256, T), T, 0, stream>>>(gbuf, G * 256);
    k_pool_max<<<blks((long)N * 256, T), T, 0, stream>>>(bufD, bat, gbuf, N);

    // ---- FC head
    k_fc1<<<blks(64 * 128, T), T, 0, stream>>>(gbuf, Wf1, bf1, g1);
    k_fc2<<<blks(640, 128), 128, 0, stream>>>(g1, Wf2, bf2, (float*)d_out);
}